// SAGEConvLayer_24068996727347
// MI455X (gfx1250) — compile-verified
//
#include <hip/hip_runtime.h>
#include <hip/hip_bf16.h>

#define N_NODES 100000
#define N_EDGES 1600000
#define D 128
#define ALPHA 0.01f
#define BN_EPS 1e-5f

typedef __attribute__((ext_vector_type(2))) float v2f;
typedef __attribute__((ext_vector_type(8))) float v8f;

// ---------------------------------------------------------------------------
// Phase 1: scatter-add x[src] rows into msg_sum[dst], count degrees.
// 32 lanes per edge; each lane handles 4 consecutive channels via float4 load
// + 4 global f32 atomics (L2-resident: msg_sum is 51MB << 192MB L2).
// ---------------------------------------------------------------------------
__global__ __launch_bounds__(256) void sage_scatter(
    const float* __restrict__ x, const long long* __restrict__ ei,
    float* __restrict__ msum, float* __restrict__ cnt) {
  long long gid = (long long)blockIdx.x * blockDim.x + threadIdx.x;
  int e = (int)(gid >> 5);
  int lane = (int)(gid & 31);
  if (e >= N_EDGES) return;
  int src = (int)ei[e];
  int dst = (int)ei[(long long)N_EDGES + e];
  const float4 v = ((const float4*)(x + (size_t)src * D))[lane];
  float* o = msum + (size_t)dst * D + lane * 4;
  atomicAdd(o + 0, v.x);
  atomicAdd(o + 1, v.y);
  atomicAdd(o + 2, v.z);
  atomicAdd(o + 3, v.w);
  if (lane == 0) atomicAdd(cnt + dst, 1.0f);
}

// ---------------------------------------------------------------------------
// Phase 2: fused GEMM  h = mean @ W_l^T + x @ W_r^T + b_l   via
// V_WMMA_F32_16X16X4_F32. One block = 16 rows x 128 cols, 8 waves, each wave
// computes one 16x16 tile. A tile (16 x 256, mean||x) staged in LDS with
// stride-257 padding (bank-conflict free). mean computed on the fly.
// ---------------------------------------------------------------------------
#define A_STRIDE 257  // 256 + 1 float pad -> distinct LDS banks across rows

__global__ __launch_bounds__(256) void sage_gemm(
    const float* __restrict__ msum, const float* __restrict__ cnt,
    const float* __restrict__ x, const float* __restrict__ Wl,
    const float* __restrict__ Wr, const float* __restrict__ bl,
    float* __restrict__ h) {
  __shared__ float As[16 * A_STRIDE];

  const int row0 = blockIdx.x * 16;
  const int tid = threadIdx.x;

  // Cooperative A-tile load: k in [0,128) = mean row, k in [128,256) = x row.
  for (int idx = tid; idx < 16 * 256; idx += 256) {
    int r = idx >> 8;
    int k = idx & 255;
    int row = row0 + r;  // N_NODES % 16 == 0 -> always in range
    float val;
    if (k < D) {
      float c = cnt[row];
      val = msum[(size_t)row * D + k] / fmaxf(c, 1.0f);
    } else {
      val = x[(size_t)row * D + (k - D)];
    }
    As[r * A_STRIDE + k] = val;
  }
  __syncthreads();

  const int wave = tid >> 5;
  const int lane = tid & 31;
  const int n = lane & 15;          // column within 16-wide tile (B/C/D layout)
  const int m = lane & 15;          // A-matrix row for this lane
  const int khi = (lane >> 4) << 1; // lanes 0-15: K {0,1}; lanes 16-31: K {2,3}
  const int col0 = wave * 16;

  v8f c = {0.f, 0.f, 0.f, 0.f, 0.f, 0.f, 0.f, 0.f};

  // K = 0..127 : mean path against W_l  (B[k][n] = W_l[n*128 + k])
  for (int kb = 0; kb < D; kb += 4) {
    v2f a, b;
    a.x = As[m * A_STRIDE + kb + khi];
    a.y = As[m * A_STRIDE + kb + khi + 1];
    b.x = Wl[(size_t)(col0 + n) * D + kb + khi];
    b.y = Wl[(size_t)(col0 + n) * D + kb + khi + 1];
    c = __builtin_amdgcn_wmma_f32_16x16x4_f32(false, a, false, b, (short)0, c,
                                              false, false);
  }
  // K = 128..255 : x path against W_r
  for (int kb = 0; kb < D; kb += 4) {
    v2f a, b;
    a.x = As[m * A_STRIDE + D + kb + khi];
    a.y = As[m * A_STRIDE + D + kb + khi + 1];
    b.x = Wr[(size_t)(col0 + n) * D + kb + khi];
    b.y = Wr[(size_t)(col0 + n) * D + kb + khi + 1];
    c = __builtin_amdgcn_wmma_f32_16x16x4_f32(false, a, false, b, (short)0, c,
                                              false, false);
  }

  // D layout: VGPR r -> M = r (lanes 0-15) / r+8 (lanes 16-31), N = lane&15.
  const float bias = bl[col0 + n];
  const int moff = (lane >> 4) * 8;
  for (int r = 0; r < 8; ++r) {
    int row = row0 + r + moff;
    h[(size_t)row * D + col0 + n] = c[r] + bias;
  }
}

// ---------------------------------------------------------------------------
// Phase 3: per-channel sum / sumsq over the node dimension.
// Thread handles channel tid&127, rows (blockIdx*2 + tid>>7) stride gridDim*2.
// LDS fold of the two row-halves, then one atomicAdd per channel per block.
// ---------------------------------------------------------------------------
__global__ __launch_bounds__(256) void sage_stats(
    const float* __restrict__ h, float* __restrict__ st) {
  const int ch = threadIdx.x & 127;
  const int half = threadIdx.x >> 7;
  float s = 0.f, q = 0.f;
  for (int row = blockIdx.x * 2 + half; row < N_NODES; row += gridDim.x * 2) {
    float v = h[(size_t)row * D + ch];
    s += v;
    q += v * v;
  }
  __shared__ float ls[256];
  __shared__ float lq[256];
  ls[threadIdx.x] = s;
  lq[threadIdx.x] = q;
  __syncthreads();
  if (threadIdx.x < 128) {
    s = ls[threadIdx.x] + ls[threadIdx.x + 128];
    q = lq[threadIdx.x] + lq[threadIdx.x + 128];
    atomicAdd(&st[ch], s);
    atomicAdd(&st[128 + ch], q);
  }
}

// ---------------------------------------------------------------------------
// Phase 4: BatchNorm (training stats, biased variance) + LeakyReLU, in place.
// ---------------------------------------------------------------------------
__global__ __launch_bounds__(256) void sage_bn_lrelu(
    float* __restrict__ h, const float* __restrict__ st,
    const float* __restrict__ gamma, const float* __restrict__ beta) {
  long long i = (long long)blockIdx.x * blockDim.x + threadIdx.x;
  if (i >= (long long)N_NODES * D) return;
  int ch = (int)(i & (D - 1));
  const float invN = 1.0f / (float)N_NODES;
  float mu = st[ch] * invN;
  float var = st[D + ch] * invN - mu * mu;
  float inv = rsqrtf(var + BN_EPS) * gamma[ch];
  float v = (h[i] - mu) * inv + beta[ch];
  h[i] = (v >= 0.f) ? v : ALPHA * v;
}

// ---------------------------------------------------------------------------
// Launcher. Workspace: [msg_sum N*128 | cnt N | stats 256] floats (~52 MB).
// h lives in d_out and is normalized in place.
// ---------------------------------------------------------------------------
extern "C" void kernel_launch(void* const* d_in, const int* in_sizes, int n_in,
                              void* d_out, int out_size, void* d_ws,
                              size_t ws_size, hipStream_t stream) {
  const float* x = (const float*)d_in[0];
  const long long* ei = (const long long*)d_in[1];  // int64 edge_idx [2, E]
  const float* Wl = (const float*)d_in[2];
  const float* bl = (const float*)d_in[3];
  const float* Wr = (const float*)d_in[4];
  const float* gamma = (const float*)d_in[5];
  const float* beta = (const float*)d_in[6];
  float* h = (float*)d_out;

  float* msum = (float*)d_ws;
  float* cnt = msum + (size_t)N_NODES * D;
  float* st = cnt + N_NODES;
  size_t zero_bytes = ((size_t)N_NODES * D + N_NODES + 256) * sizeof(float);

  hipMemsetAsync(d_ws, 0, zero_bytes, stream);

  {  // scatter: 32 lanes per edge
    long long threads = (long long)N_EDGES * 32;
    int blocks = (int)((threads + 255) / 256);
    sage_scatter<<<blocks, 256, 0, stream>>>(x, ei, msum, cnt);
  }
  {  // fused WMMA GEMM: 16 rows per block
    sage_gemm<<<N_NODES / 16, 256, 0, stream>>>(msum, cnt, x, Wl, Wr, bl, h);
  }
  {  // channel stats
    sage_stats<<<1024, 256, 0, stream>>>(h, st);
  }
  {  // BN + LeakyReLU in place
    long long total = (long long)N_NODES * D;
    int blocks = (int)((total + 255) / 256);
    sage_bn_lrelu<<<blocks, 256, 0, stream>>>(h, st, gamma, beta);
  }
}